// CrossCluster_54795192762833
// MI455X (gfx1250) — compile-verified
//
#include <hip/hip_runtime.h>
#include <math.h>

// ---------------------------------------------------------------------------
// CrossCluster for MI455X (gfx1250, wave32, WMMA).
// fp32 end-to-end; matrix work on V_WMMA_F32_16X16X4_F32.
// HBM traffic ~133MB -> ~5.7us @ 23.3 TB/s; 16 GFLOP fp32 via WMMA.
// Round-2 change: A-fragment register reuse across 4 N-tiles (pass1) / 2
// N-tiles (pass2) to cut L2 weight traffic 4x/2x below the HBM roofline cost.
// ---------------------------------------------------------------------------

typedef float v2f __attribute__((ext_vector_type(2)));
typedef float v8f __attribute__((ext_vector_type(8)));

#define NSP 32768   // 32*32*32 spatial points
// workspace layout (float offsets)
#define WS_W      0         // packed [512][96] f_mri_w|v_mri_w
#define WS_BIAS   49152     // [512]
#define WS_POOL   49664     // [2][96][8]
#define WS_CENT   51200     // [16][8][32] normalized f-centers
#define WS_VCENT  55296     // [16][8][32] raw v-centers
#define WS_AGGN   59392     // [16][8][32] atomic numerator
#define WS_DEN    63488     // [16][8]     atomic denominator
#define WS_AGG    63616     // [16][8][32] finalized agg
#define WS_SIM    67712     // [16][32768] top-1 sim value
#define WS_IDX    592000    // [16][32768] top-1 center index (u32)

// ---------------------------------------------------------------------------
// K0: pack MRI-side weights into one 512-row GEMM A-matrix; zero accumulators.
__global__ __launch_bounds__(256) void k_pack(const float* __restrict__ fw,
                                              const float* __restrict__ fb,
                                              const float* __restrict__ vw,
                                              const float* __restrict__ vb,
                                              float* __restrict__ ws) {
  int i = blockIdx.x * 256 + threadIdx.x;
  if (i < 512 * 96) {
    int row = i / 96;
    ws[WS_W + i] = (row < 256) ? fw[i] : vw[i - 256 * 96];
  }
  if (i < 512)  ws[WS_BIAS + i] = (i < 256) ? fb[i] : vb[i - 256];
  if (i < 4096) ws[WS_AGGN + i] = 0.f;
  if (i < 128)  ws[WS_DEN  + i] = 0.f;
}

// ---------------------------------------------------------------------------
// K1: AdaptiveAvgPool3d((2,2,2)) of PET: mean over 16x16x16 blocks.
__global__ __launch_bounds__(256) void k_pool(const float* __restrict__ PET,
                                              float* __restrict__ ws) {
  __shared__ float red[256];
  int blk = blockIdx.x;            // (b*96+c)*8 + m
  int m  = blk & 7;
  int bc = blk >> 3;
  int pw = m >> 2, ph = (m >> 1) & 1, pd = m & 1;
  const float* src = PET + (size_t)bc * NSP;
  float s = 0.f;
  for (int j = 0; j < 16; ++j) {
    int l = threadIdx.x + j * 256;                 // 0..4095 within block
    int iw = l >> 8, ih = (l >> 4) & 15, id = l & 15;
    s += src[((pw * 16 + iw) * 32 + (ph * 16 + ih)) * 32 + (pd * 16 + id)];
  }
  red[threadIdx.x] = s;
  __syncthreads();
  for (int off = 128; off > 0; off >>= 1) {
    if (threadIdx.x < off) red[threadIdx.x] += red[threadIdx.x + off];
    __syncthreads();
  }
  if (threadIdx.x == 0) ws[WS_POOL + blk] = red[0] * (1.f / 4096.f);
}

// ---------------------------------------------------------------------------
// K2: centers = f_pet conv of pooled PET (pool/conv commute), L2-normalized;
// v_centers raw. One wave32 per (bp, m) pair, lane = channel c.
__global__ __launch_bounds__(32) void k_centers(const float* __restrict__ fpw,
                                                const float* __restrict__ fpb,
                                                const float* __restrict__ vpw,
                                                const float* __restrict__ vpb,
                                                float* __restrict__ ws) {
  int pair = blockIdx.x;           // bp*8 + m
  int m = pair & 7, bp = pair >> 3;
  int b = bp >> 3,  h = bp & 7;
  int c = threadIdx.x;             // 0..31
  int o = h * 32 + c;
  const float* pool = ws + WS_POOL;
  float f = fpb[o], v = vpb[o];
  for (int k = 0; k < 96; ++k) {
    float p = pool[(b * 96 + k) * 8 + m];
    f += fpw[o * 96 + k] * p;
    v += vpw[o * 96 + k] * p;
  }
  float ss = f * f;
  for (int off = 16; off > 0; off >>= 1) ss += __shfl_xor(ss, off, 32);
  float inv = 1.f / fmaxf(sqrtf(ss), 1e-12f);
  ws[WS_CENT  + pair * 32 + c] = f * inv;
  ws[WS_VCENT + pair * 32 + c] = v;
}

// ---------------------------------------------------------------------------
// K3: fused [512x96]@[96x64]-per-group WMMA GEMM (fM rows 0..255, vM rows
// 256..511) + per-point norm/cos-sim/sigmoid/argmax + agg accumulation.
// Block: 256 threads (8 waves); each wave owns 4 M-tiles x 4 N-tiles
// (A fragments reused across the 4 N-tiles); 4 groups = 256 points/block.
__global__ __launch_bounds__(256) void k_pass1(const float* __restrict__ MRI,
                                               const float* __restrict__ alpha_p,
                                               const float* __restrict__ beta_p,
                                               float* __restrict__ ws,
                                               unsigned int* __restrict__ wsidx) {
  __shared__ float sC[512 * 16];   // staged C tile (32KB), one N-tile at a time
  __shared__ float sCent[2048];    // normalized centers for this batch
  __shared__ float sAgg[2048];     // block-local agg numerator
  __shared__ float sDen[64];       // block-local agg denominator
  int b      = blockIdx.x >> 7;
  int n0base = (blockIdx.x & 127) * 256;
  int tid = threadIdx.x, lane = tid & 31, wv = tid >> 5;
  int lhi = lane >> 4, l15 = lane & 15;
  const float* W    = ws + WS_W;
  const float* bias = ws + WS_BIAS;
  float alpha = alpha_p[0], beta = beta_p[0];

  for (int i = tid; i < 2048; i += 256) {
    sCent[i] = ws[WS_CENT + b * 2048 + i];
    sAgg[i]  = 0.f;
  }
  if (tid < 64) sDen[tid] = 0.f;
  __syncthreads();

  const float* Mb = MRI + (size_t)b * 96 * NSP;

  for (int g = 0; g < 4; ++g) {
    int n0g = n0base + g * 64;
    v8f acc[4][4] = {};            // [mtile][ntile], 128 VGPRs
    for (int k = 0; k < 96; k += 4) {
      int kk = k + lhi * 2;        // A/B: lanes0-15 K=k,k+1; lanes16-31 K=k+2,k+3
      v2f bf[4];
#pragma unroll
      for (int nt = 0; nt < 4; ++nt) {
        bf[nt].x = Mb[(size_t)kk * NSP + n0g + nt * 16 + l15];
        bf[nt].y = Mb[(size_t)(kk + 1) * NSP + n0g + nt * 16 + l15];
      }
#pragma unroll
      for (int mt = 0; mt < 4; ++mt) {
        int row = wv * 64 + mt * 16 + l15;   // A-matrix M = lane&15
        v2f af;
        af.x = W[row * 96 + kk];
        af.y = W[row * 96 + kk + 1];
#pragma unroll
        for (int nt = 0; nt < 4; ++nt)       // A reused 4x from registers
          acc[mt][nt] = __builtin_amdgcn_wmma_f32_16x16x4_f32(
              false, af, false, bf[nt], (short)0, acc[mt][nt], false, false);
      }
    }

#pragma unroll
    for (int nt = 0; nt < 4; ++nt) {
      int n0 = n0g + nt * 16;
      __syncthreads();             // previous tile's consumers done with sC
#pragma unroll
      for (int mt = 0; mt < 4; ++mt)
#pragma unroll
        for (int r = 0; r < 8; ++r) {
          int row = wv * 64 + mt * 16 + r + lhi * 8;  // C: VGPR r -> M=r / r+8
          sC[row * 16 + l15] = acc[mt][nt][r] + bias[row];
        }
      __syncthreads();

      // per-(point,head) post-processing: 128 pairs x 2 threads (16 ch each)
      int pairid = tid >> 1, half = tid & 1;
      int h = pairid >> 4, pt = pairid & 15;
      int cbase = half * 16;
      float fv[16], vv[16];
      float ssq = 0.f;
#pragma unroll
      for (int j = 0; j < 16; ++j) {
        int c = cbase + j;
        fv[j] = sC[(h * 32 + c) * 16 + pt];
        vv[j] = sC[(256 + h * 32 + c) * 16 + pt];
        ssq += fv[j] * fv[j];
      }
      ssq += __shfl_xor(ssq, 1, 32);            // partner thread in-wave
      float inv = 1.f / fmaxf(sqrtf(ssq), 1e-12f);
      float best = -1.f;
      int bestm = 0;
#pragma unroll
      for (int m = 0; m < 8; ++m) {
        float d = 0.f;
#pragma unroll
        for (int j = 0; j < 16; ++j) d += fv[j] * sCent[(h * 8 + m) * 32 + cbase + j];
        d += __shfl_xor(d, 1, 32);
        float sim = 1.f / (1.f + __expf(-(beta + alpha * d * inv)));
        if (sim > best) { best = sim; bestm = m; }  // strict > == first-max
      }
#pragma unroll
      for (int j = 0; j < 16; ++j)
        atomicAdd(&sAgg[(h * 8 + bestm) * 32 + cbase + j], best * vv[j]);
      if (half == 0) {
        atomicAdd(&sDen[h * 8 + bestm], best);
        int n = n0 + pt;
        ws[WS_SIM + (b * 8 + h) * NSP + n] = best;
        wsidx[(b * 8 + h) * NSP + n]       = (unsigned int)bestm;
      }
    }
    __syncthreads();
  }

  // flush block-local accumulators (2112 global f32 atomics per block)
  for (int i = tid; i < 2048; i += 256)
    unsafeAtomicAdd(ws + WS_AGGN + b * 2048 + i, sAgg[i]);
  if (tid < 64)
    unsafeAtomicAdd(ws + WS_DEN + b * 64 + tid, sDen[tid]);
}

// ---------------------------------------------------------------------------
// K4: agg = (aggnum + v_centers) / (den + 1)
__global__ __launch_bounds__(256) void k_finalize(float* __restrict__ ws) {
  int i = blockIdx.x * 256 + threadIdx.x;
  if (i < 4096) {
    int bpm = i >> 5;
    ws[WS_AGG + i] = (ws[WS_AGGN + i] + ws[WS_VCENT + i]) / (ws[WS_DEN + bpm] + 1.f);
  }
}

// ---------------------------------------------------------------------------
// K5: dispatch + proj. Build B[256x32] = agg[h,idx(n),c]*sim(n) in LDS, then
// [96x256]@[256x32] WMMA (6 waves x 1 M-tile x 2 N-tiles, A reused across the
// N-tiles), 2 groups per block = 64 points.
__global__ __launch_bounds__(192) void k_pass2(const float* __restrict__ projw,
                                               const float* __restrict__ projb,
                                               const float* __restrict__ ws,
                                               const unsigned int* __restrict__ wsidx,
                                               float* __restrict__ out) {
  __shared__ float sB[256 * 32];   // 32KB
  __shared__ float sAgg[2048];     // 8KB
  int b      = blockIdx.x >> 9;
  int n0base = (blockIdx.x & 511) * 64;
  int tid = threadIdx.x, lane = tid & 31, wv = tid >> 5;
  int lhi = lane >> 4, l15 = lane & 15;

  for (int i = tid; i < 2048; i += 192) sAgg[i] = ws[WS_AGG + b * 2048 + i];
  __syncthreads();

  for (int s = 0; s < 2; ++s) {
    int n0 = n0base + s * 32;
    for (int i = tid; i < 8192; i += 192) {
      int c256 = i >> 5, pt = i & 31;            // sB[c256][pt], pt = local col
      int h = c256 >> 5, c = c256 & 31;
      int n = n0 + pt;
      unsigned int idx = wsidx[(b * 8 + h) * NSP + n];
      float sim        = ws[WS_SIM + (b * 8 + h) * NSP + n];
      sB[i] = sAgg[h * 256 + idx * 32 + c] * sim;
    }
    __syncthreads();

    v8f acc[2] = {};
    for (int k = 0; k < 256; k += 4) {
      int kk = k + lhi * 2;
      int row = wv * 16 + l15;
      v2f af;
      af.x = projw[row * 256 + kk];
      af.y = projw[row * 256 + kk + 1];
#pragma unroll
      for (int nt = 0; nt < 2; ++nt) {
        v2f bfr;
        bfr.x = sB[kk * 32 + nt * 16 + l15];
        bfr.y = sB[(kk + 1) * 32 + nt * 16 + l15];
        acc[nt] = __builtin_amdgcn_wmma_f32_16x16x4_f32(
            false, af, false, bfr, (short)0, acc[nt], false, false);
      }
    }
#pragma unroll
    for (int nt = 0; nt < 2; ++nt)
#pragma unroll
      for (int r = 0; r < 8; ++r) {
        int row = wv * 16 + r + lhi * 8;
        out[((size_t)b * 96 + row) * NSP + n0 + nt * 16 + l15] = acc[nt][r] + projb[row];
      }
    __syncthreads();
  }
}

// ---------------------------------------------------------------------------
extern "C" void kernel_launch(void* const* d_in, const int* in_sizes, int n_in,
                              void* d_out, int out_size, void* d_ws, size_t ws_size,
                              hipStream_t stream) {
  const float* PET     = (const float*)d_in[0];
  const float* MRI     = (const float*)d_in[1];
  const float* f_pet_w = (const float*)d_in[2];
  const float* f_pet_b = (const float*)d_in[3];
  const float* f_mri_w = (const float*)d_in[4];
  const float* f_mri_b = (const float*)d_in[5];
  const float* v_pet_w = (const float*)d_in[6];
  const float* v_pet_b = (const float*)d_in[7];
  const float* v_mri_w = (const float*)d_in[8];
  const float* v_mri_b = (const float*)d_in[9];
  const float* proj_w  = (const float*)d_in[10];
  const float* proj_b  = (const float*)d_in[11];
  const float* alpha   = (const float*)d_in[12];
  const float* beta    = (const float*)d_in[13];
  float* ws = (float*)d_ws;
  unsigned int* wsidx = (unsigned int*)(ws + WS_IDX);
  float* out = (float*)d_out;

  hipLaunchKernelGGL(k_pack,     dim3(192),  dim3(256), 0, stream,
                     f_mri_w, f_mri_b, v_mri_w, v_mri_b, ws);
  hipLaunchKernelGGL(k_pool,     dim3(1536), dim3(256), 0, stream, PET, ws);
  hipLaunchKernelGGL(k_centers,  dim3(128),  dim3(32),  0, stream,
                     f_pet_w, f_pet_b, v_pet_w, v_pet_b, ws);
  hipLaunchKernelGGL(k_pass1,    dim3(256),  dim3(256), 0, stream,
                     MRI, alpha, beta, ws, wsidx);
  hipLaunchKernelGGL(k_finalize, dim3(16),   dim3(256), 0, stream, ws);
  hipLaunchKernelGGL(k_pass2,    dim3(1024), dim3(192), 0, stream,
                     proj_w, proj_b, ws, wsidx, out);
}